// ComplexCNNWithoutDropout_90855738179625
// MI455X (gfx1250) — compile-verified
//
#include <hip/hip_runtime.h>
#include <math.h>

// ---------------------------------------------------------------------------
// Complex CNN forward (fp32), CDNA5/gfx1250.
// Linear layers: split-K complex GEMM on V_WMMA_F32_16X16X4_F32, b128 loads.
// Workspace requirement: ~224 MB.
// ---------------------------------------------------------------------------

typedef float v2f __attribute__((ext_vector_type(2)));
typedef float v4f __attribute__((ext_vector_type(4)));
typedef float v8f __attribute__((ext_vector_type(8)));

namespace {
constexpr int B_   = 64;
constexpr int H1   = 116, W1 = 212;
constexpr int C1_  = 10,  C2_ = 20;
constexpr int H1O  = 112, W1O = 208;   // conv1 out
constexpr int P1H  = 56,  P1W = 104;   // pool1 out
constexpr int H2O  = 52,  W2O = 100;   // conv2 out
constexpr int P2H  = 26,  P2W = 50;    // pool2 out
constexpr int FLAT_ = C2_ * P2H * P2W; // 26000
constexpr int HID_  = 512, NCLS_ = 10;
constexpr float EPS_ = 1e-5f;
}

// ---------------------------------------------------------------------------
// conv1: [B,1,116,212] (r,i) -> [B,10,112,208] (r,i), 5x5 VALID, complex.
// ---------------------------------------------------------------------------
__global__ void conv1_kernel(const float* __restrict__ xr, const float* __restrict__ xi,
                             const float* __restrict__ wr, const float* __restrict__ wi,
                             const float* __restrict__ br, const float* __restrict__ bi,
                             float* __restrict__ yr, float* __restrict__ yi) {
    __shared__ float swr[C1_ * 25];
    __shared__ float swi[C1_ * 25];
    for (int t = threadIdx.x; t < C1_ * 25; t += blockDim.x) { swr[t] = wr[t]; swi[t] = wi[t]; }
    __syncthreads();

    int idx = blockIdx.x * blockDim.x + threadIdx.x;
    const int total = B_ * C1_ * H1O * W1O;
    if (idx >= total) return;
    int w = idx % W1O; int t = idx / W1O;
    int h = t % H1O;   t /= H1O;
    int c = t % C1_;   int b = t / C1_;

    const float* xrb = xr + (long)b * H1 * W1;
    const float* xib = xi + (long)b * H1 * W1;
    float accr = br[c], acci = bi[c];
#pragma unroll
    for (int kh = 0; kh < 5; ++kh) {
        const float* xr0 = xrb + (h + kh) * W1 + w;
        const float* xi0 = xib + (h + kh) * W1 + w;
#pragma unroll
        for (int kw = 0; kw < 5; ++kw) {
            float ar = xr0[kw], ai = xi0[kw];
            float wre = swr[c * 25 + kh * 5 + kw];
            float wim = swi[c * 25 + kh * 5 + kw];
            accr = fmaf(ar, wre, accr); accr = fmaf(-ai, wim, accr);
            acci = fmaf(ar, wim, acci); acci = fmaf(ai, wre, acci);
        }
    }
    yr[idx] = accr; yi[idx] = acci;
}

// ---------------------------------------------------------------------------
// conv2: [B,10,56,104] -> [B,20,52,100], 5x5 VALID, complex. Weights in LDS.
// ---------------------------------------------------------------------------
__global__ void conv2_kernel(const float* __restrict__ xr, const float* __restrict__ xi,
                             const float* __restrict__ wr, const float* __restrict__ wi,
                             const float* __restrict__ br, const float* __restrict__ bi,
                             float* __restrict__ yr, float* __restrict__ yi) {
    __shared__ float swr[C2_ * C1_ * 25];   // 20 KB
    __shared__ float swi[C2_ * C1_ * 25];   // 20 KB (LDS is 320 KB/WGP on CDNA5)
    for (int t = threadIdx.x; t < C2_ * C1_ * 25; t += blockDim.x) { swr[t] = wr[t]; swi[t] = wi[t]; }
    __syncthreads();

    int idx = blockIdx.x * blockDim.x + threadIdx.x;
    const int total = B_ * C2_ * H2O * W2O;
    if (idx >= total) return;
    int w = idx % W2O; int t = idx / W2O;
    int h = t % H2O;   t /= H2O;
    int c = t % C2_;   int b = t / C2_;

    float accr = br[c], acci = bi[c];
    for (int ci = 0; ci < C1_; ++ci) {
        const float* xrb = xr + ((long)(b * C1_ + ci) * P1H + h) * P1W + w;
        const float* xib = xi + ((long)(b * C1_ + ci) * P1H + h) * P1W + w;
        const float* wrc = &swr[(c * C1_ + ci) * 25];
        const float* wic = &swi[(c * C1_ + ci) * 25];
#pragma unroll
        for (int kh = 0; kh < 5; ++kh) {
#pragma unroll
            for (int kw = 0; kw < 5; ++kw) {
                float ar = xrb[kh * P1W + kw], ai = xib[kh * P1W + kw];
                float wre = wrc[kh * 5 + kw], wim = wic[kh * 5 + kw];
                accr = fmaf(ar, wre, accr); accr = fmaf(-ai, wim, accr);
                acci = fmaf(ar, wim, acci); acci = fmaf(ai, wre, acci);
            }
        }
    }
    yr[idx] = accr; yi[idx] = acci;
}

// ---------------------------------------------------------------------------
// zero a float buffer (stats must be re-zeroed every call: ws is poisoned)
// ---------------------------------------------------------------------------
__global__ void zero_kernel(float* __restrict__ p, int n) {
    int i = blockIdx.x * blockDim.x + threadIdx.x;
    if (i < n) p[i] = 0.0f;
}

// ---------------------------------------------------------------------------
// per-channel 5-moment reduction over (B,H,W) of [B,C,H,W] r/i pair.
// stats[c*5 + {0..4}] = { Sr, Si, Srr, Sii, Sri }
// ---------------------------------------------------------------------------
__global__ void moments_conv_kernel(const float* __restrict__ r, const float* __restrict__ im,
                                    int C, int HW, float* __restrict__ stats) {
    const int c   = blockIdx.x;
    const int tid = threadIdx.x;
    const long N  = (long)B_ * HW;
    float s0 = 0.f, s1 = 0.f, s2 = 0.f, s3 = 0.f, s4 = 0.f;
    for (long idx = (long)blockIdx.y * blockDim.x + tid; idx < N;
         idx += (long)gridDim.y * blockDim.x) {
        long bb = idx / HW, sp = idx % HW;
        long off = ((bb * C) + c) * (long)HW + sp;
        float a = r[off], b = im[off];
        s0 += a; s1 += b; s2 += a * a; s3 += b * b; s4 += a * b;
    }
    __shared__ float red[5][256];
    red[0][tid] = s0; red[1][tid] = s1; red[2][tid] = s2; red[3][tid] = s3; red[4][tid] = s4;
    __syncthreads();
    for (int s = blockDim.x >> 1; s > 0; s >>= 1) {
        if (tid < s) {
#pragma unroll
            for (int j = 0; j < 5; ++j) red[j][tid] += red[j][tid + s];
        }
        __syncthreads();
    }
    if (tid == 0) {
#pragma unroll
        for (int j = 0; j < 5; ++j) atomicAdd(&stats[c * 5 + j], red[j][0]);
    }
}

// per-feature moments over batch for FC layers ([B,F] layout); full overwrite.
__global__ void moments_fc_kernel(const float* __restrict__ r, const float* __restrict__ im,
                                  int F, float* __restrict__ stats) {
    int f = blockIdx.x * blockDim.x + threadIdx.x;
    if (f >= F) return;
    float s0 = 0.f, s1 = 0.f, s2 = 0.f, s3 = 0.f, s4 = 0.f;
    for (int b = 0; b < B_; ++b) {
        float a = r[(long)b * F + f], c = im[(long)b * F + f];
        s0 += a; s1 += c; s2 += a * a; s3 += c * c; s4 += a * c;
    }
    stats[f * 5 + 0] = s0; stats[f * 5 + 1] = s1; stats[f * 5 + 2] = s2;
    stats[f * 5 + 3] = s3; stats[f * 5 + 4] = s4;
}

// ---------------------------------------------------------------------------
// fold complex-BN whitening + gamma into per-channel affine:
// coef[c*8+{0..7}] = { mr, mi, Arr, Ari, Air, Aii, ber, bei }
// ---------------------------------------------------------------------------
__global__ void bncoef_kernel(const float* __restrict__ stats,
                              const float* __restrict__ grr, const float* __restrict__ gri,
                              const float* __restrict__ gii,
                              const float* __restrict__ ber, const float* __restrict__ bei,
                              float invN, int C, float* __restrict__ coef) {
    int c = blockIdx.x * blockDim.x + threadIdx.x;
    if (c >= C) return;
    float mr  = stats[c * 5 + 0] * invN;
    float mi  = stats[c * 5 + 1] * invN;
    float Vrr = stats[c * 5 + 2] * invN - mr * mr + EPS_;
    float Vii = stats[c * 5 + 3] * invN - mi * mi + EPS_;
    float Vri = stats[c * 5 + 4] * invN - mr * mi;
    float s   = sqrtf(Vrr * Vii - Vri * Vri);
    float t   = sqrtf(Vrr + Vii + 2.0f * s);
    float inv = 1.0f / (s * t);
    float Rrr = (Vii + s) * inv, Rii = (Vrr + s) * inv, Rri = -Vri * inv;
    float a = grr[c], g2 = gri[c], g3 = gii[c];
    coef[c * 8 + 0] = mr;
    coef[c * 8 + 1] = mi;
    coef[c * 8 + 2] = a * Rrr + g2 * Rri;   // Arr
    coef[c * 8 + 3] = a * Rri + g2 * Rii;   // Ari
    coef[c * 8 + 4] = g2 * Rrr + g3 * Rri;  // Air
    coef[c * 8 + 5] = g2 * Rri + g3 * Rii;  // Aii
    coef[c * 8 + 6] = ber[c];
    coef[c * 8 + 7] = bei[c];
}

// ---------------------------------------------------------------------------
// fused BN-affine + CReLU + 2x2 magnitude-argmax pool ([B,C,H,W]->[B,C,H/2,W/2])
// scan order (dh,dw) = (0,0),(0,1),(1,0),(1,1) with strict > == argmax first-max
// ---------------------------------------------------------------------------
__global__ void bnpool_kernel(const float* __restrict__ r, const float* __restrict__ im,
                              const float* __restrict__ coef, int C, int H, int W,
                              float* __restrict__ pr, float* __restrict__ pi) {
    const int HO = H >> 1, WO = W >> 1;
    const int total = B_ * C * HO * WO;
    int idx = blockIdx.x * blockDim.x + threadIdx.x;
    if (idx >= total) return;
    int w = idx % WO; int t = idx / WO;
    int h = t % HO;   t /= HO;
    int c = t % C;    int b = t / C;

    float mr  = coef[c * 8 + 0], mi  = coef[c * 8 + 1];
    float Arr = coef[c * 8 + 2], Ari = coef[c * 8 + 3];
    float Air = coef[c * 8 + 4], Aii = coef[c * 8 + 5];
    float br  = coef[c * 8 + 6], bi  = coef[c * 8 + 7];

    long base = (((long)b * C + c) * H + 2 * h) * W + 2 * w;
    float bestR = 0.f, bestI = 0.f, bestM = -1.f;
#pragma unroll
    for (int dh = 0; dh < 2; ++dh) {
#pragma unroll
        for (int dw = 0; dw < 2; ++dw) {
            long off = base + dh * W + dw;
            float cr = r[off] - mr, ci = im[off] - mi;
            float orr = fmaxf(fmaf(Arr, cr, fmaf(Ari, ci, br)), 0.0f);
            float oii = fmaxf(fmaf(Air, cr, fmaf(Aii, ci, bi)), 0.0f);
            float m = orr * orr + oii * oii;
            if (m > bestM) { bestM = m; bestR = orr; bestI = oii; }
        }
    }
    pr[idx] = bestR; pi[idx] = bestI;
}

// fused BN-affine + CReLU for FC layers ([B,F]), in-place safe (own element).
__global__ void bnrelu_fc_kernel(float* __restrict__ r, float* __restrict__ im,
                                 const float* __restrict__ coef, int F) {
    int idx = blockIdx.x * blockDim.x + threadIdx.x;
    if (idx >= B_ * F) return;
    int f = idx % F;
    float mr  = coef[f * 8 + 0], mi  = coef[f * 8 + 1];
    float Arr = coef[f * 8 + 2], Ari = coef[f * 8 + 3];
    float Air = coef[f * 8 + 4], Aii = coef[f * 8 + 5];
    float br  = coef[f * 8 + 6], bi  = coef[f * 8 + 7];
    float cr = r[idx] - mr, ci = im[idx] - mi;
    r[idx]  = fmaxf(fmaf(Arr, cr, fmaf(Ari, ci, br)), 0.0f);
    im[idx] = fmaxf(fmaf(Air, cr, fmaf(Aii, ci, bi)), 0.0f);
}

// ---------------------------------------------------------------------------
// Split-K complex GEMM via V_WMMA_F32_16X16X4_F32.
//   partial[s] += Xr*Wr^T - Xi*Wi^T  (real) / Xr*Wi^T + Xi*Wr^T (imag)
// X:[M,K], W:[N,K] row-major. One wave -> one 16x16 tile of one K-chunk.
//
// K-slot permutation trick: WMMA sums over its 4 K-slots; the reduction is
// order-independent, so any assignment of K indices to (component, lane-half)
// slots is valid as long as A and B use the SAME assignment. Each lane loads a
// float4 (global_load_b128) at [k + 4*half]; super-step K=8 issues two WMMAs
// fed by the lo/hi float2 halves. A-slot map per ISA: comp v, half h -> K=v+2h;
// B mirrors it, so pairing stays consistent under the permutation.
// ---------------------------------------------------------------------------
__global__ void cgemm_splitk_wmma_kernel(const float* __restrict__ Xr, const float* __restrict__ Xi,
                                         const float* __restrict__ Wr, const float* __restrict__ Wi,
                                         float* __restrict__ PR, float* __restrict__ PI,
                                         int M, int N, int K, int S, int KC) {
    const int wid  = (blockIdx.x * blockDim.x + threadIdx.x) >> 5;
    const int lane = threadIdx.x & 31;
    const int mtiles = (M + 15) >> 4;
    const int ntiles = (N + 15) >> 4;
    const int tiles  = mtiles * ntiles;
    if (wid >= tiles * S) return;             // uniform per wave -> EXEC stays full
    const int tile = wid % tiles;
    const int s    = wid / tiles;
    const int mt = (tile % mtiles) << 4;
    const int nt = (tile / mtiles) << 4;
    const int k0 = s * KC;
    const int k1 = (k0 + KC < K) ? (k0 + KC) : K;   // KC mult of 8, K mult of 8

    const int lm = lane & 15;
    const int hb = lane >> 4;                 // lane half
    const int arow = mt + lm;                 // M is a multiple of 16 here
    const int bcol = nt + lm;
    const int brow = (bcol < N) ? bcol : 0;   // clamp: garbage feeds only masked cols

    const float* xr0 = Xr + (long)arow * K + 4 * hb;
    const float* xi0 = Xi + (long)arow * K + 4 * hb;
    const float* wr0 = Wr + (long)brow * K + 4 * hb;
    const float* wi0 = Wi + (long)brow * K + 4 * hb;

    v8f accR = {};
    v8f accI = {};
    for (int k = k0; k < k1; k += 8) {
        __builtin_prefetch(wr0 + k + 512, 0, 2);   // global_prefetch_b8, ~2KB ahead
        __builtin_prefetch(wi0 + k + 512, 0, 2);
        v4f ar4 = *(const v4f*)(xr0 + k);          // global_load_b128
        v4f ai4 = *(const v4f*)(xi0 + k);
        v4f br4 = *(const v4f*)(wr0 + k);
        v4f bi4 = *(const v4f*)(wi0 + k);
        v2f arL = __builtin_shufflevector(ar4, ar4, 0, 1);
        v2f arH = __builtin_shufflevector(ar4, ar4, 2, 3);
        v2f aiL = __builtin_shufflevector(ai4, ai4, 0, 1);
        v2f aiH = __builtin_shufflevector(ai4, ai4, 2, 3);
        v2f brL = __builtin_shufflevector(br4, br4, 0, 1);
        v2f brH = __builtin_shufflevector(br4, br4, 2, 3);
        v2f biL = __builtin_shufflevector(bi4, bi4, 0, 1);
        v2f biH = __builtin_shufflevector(bi4, bi4, 2, 3);
        v2f naiL = -aiL;
        v2f naiH = -aiH;
        // f32 WMMA has no A/B sign modifier (NEG = CNeg only), negate Ai in VALU
        accR = __builtin_amdgcn_wmma_f32_16x16x4_f32(false, arL,  false, brL, (short)0, accR, false, false);
        accR = __builtin_amdgcn_wmma_f32_16x16x4_f32(false, naiL, false, biL, (short)0, accR, false, false);
        accI = __builtin_amdgcn_wmma_f32_16x16x4_f32(false, arL,  false, biL, (short)0, accI, false, false);
        accI = __builtin_amdgcn_wmma_f32_16x16x4_f32(false, aiL,  false, brL, (short)0, accI, false, false);
        accR = __builtin_amdgcn_wmma_f32_16x16x4_f32(false, arH,  false, brH, (short)0, accR, false, false);
        accR = __builtin_amdgcn_wmma_f32_16x16x4_f32(false, naiH, false, biH, (short)0, accR, false, false);
        accI = __builtin_amdgcn_wmma_f32_16x16x4_f32(false, arH,  false, biH, (short)0, accI, false, false);
        accI = __builtin_amdgcn_wmma_f32_16x16x4_f32(false, aiH,  false, brH, (short)0, accI, false, false);
    }

    // C/D layout: VGPR v, lanes 0-15 -> row mt+v; lanes 16-31 -> row mt+v+8.
    const int col = nt + lm;
    if (col < N) {
        float* pr = PR + (long)s * M * N;
        float* pi = PI + (long)s * M * N;
        const int rbase = mt + (hb << 3);
#pragma unroll
        for (int v = 0; v < 8; ++v) {
            pr[(long)(rbase + v) * N + col] = accR[v];
            pi[(long)(rbase + v) * N + col] = accI[v];
        }
    }
}

// sum split-K partials + bias -> Y
__global__ void combine_kernel(const float* __restrict__ PR, const float* __restrict__ PI,
                               const float* __restrict__ br, const float* __restrict__ bi,
                               float* __restrict__ Yr, float* __restrict__ Yi,
                               int M, int N, int S) {
    int idx = blockIdx.x * blockDim.x + threadIdx.x;
    if (idx >= M * N) return;
    int n = idx % N;
    float sr = br[n], si = bi[n];
    for (int s = 0; s < S; ++s) {
        sr += PR[(long)s * M * N + idx];
        si += PI[(long)s * M * N + idx];
    }
    Yr[idx] = sr; Yi[idx] = si;
}

// ---------------------------------------------------------------------------
// out = log_softmax(hr^2 + hi^2) along the 10-class axis.
// ---------------------------------------------------------------------------
__global__ void logsoftmax_kernel(const float* __restrict__ hr, const float* __restrict__ hi,
                                  float* __restrict__ out) {
    int b = blockIdx.x * blockDim.x + threadIdx.x;
    if (b >= B_) return;
    float v[NCLS_];
    float mx = -INFINITY;
#pragma unroll
    for (int j = 0; j < NCLS_; ++j) {
        float a = hr[b * NCLS_ + j], c = hi[b * NCLS_ + j];
        v[j] = a * a + c * c;
        mx = fmaxf(mx, v[j]);
    }
    float s = 0.f;
#pragma unroll
    for (int j = 0; j < NCLS_; ++j) s += expf(v[j] - mx);
    float l = mx + logf(s);
#pragma unroll
    for (int j = 0; j < NCLS_; ++j) out[b * NCLS_ + j] = v[j] - l;
}

// ---------------------------------------------------------------------------
extern "C" void kernel_launch(void* const* d_in, const int* in_sizes, int n_in,
                              void* d_out, int out_size, void* d_ws, size_t ws_size,
                              hipStream_t stream) {
    // --- inputs, in setup_inputs() order ---
    const float* x_re = (const float*)d_in[0];
    const float* x_im = (const float*)d_in[1];
    const float* w1r = (const float*)d_in[2];  const float* w1i = (const float*)d_in[3];
    const float* b1r = (const float*)d_in[4];  const float* b1i = (const float*)d_in[5];
    const float* g1rr = (const float*)d_in[6]; const float* g1ri = (const float*)d_in[7];
    const float* g1ii = (const float*)d_in[8]; const float* be1r = (const float*)d_in[9];
    const float* be1i = (const float*)d_in[10];
    const float* w2r = (const float*)d_in[11]; const float* w2i = (const float*)d_in[12];
    const float* b2r = (const float*)d_in[13]; const float* b2i = (const float*)d_in[14];
    const float* g2rr = (const float*)d_in[15]; const float* g2ri = (const float*)d_in[16];
    const float* g2ii = (const float*)d_in[17]; const float* be2r = (const float*)d_in[18];
    const float* be2i = (const float*)d_in[19];
    const float* f1wr = (const float*)d_in[20]; const float* f1wi = (const float*)d_in[21];
    const float* f1br = (const float*)d_in[22]; const float* f1bi = (const float*)d_in[23];
    const float* g3rr = (const float*)d_in[24]; const float* g3ri = (const float*)d_in[25];
    const float* g3ii = (const float*)d_in[26]; const float* be3r = (const float*)d_in[27];
    const float* be3i = (const float*)d_in[28];
    const float* f2wr = (const float*)d_in[29]; const float* f2wi = (const float*)d_in[30];
    const float* f2br = (const float*)d_in[31]; const float* f2bi = (const float*)d_in[32];
    const float* g4rr = (const float*)d_in[33]; const float* g4ri = (const float*)d_in[34];
    const float* g4ii = (const float*)d_in[35]; const float* be4r = (const float*)d_in[36];
    const float* be4i = (const float*)d_in[37];
    const float* cwr = (const float*)d_in[38]; const float* cwi = (const float*)d_in[39];
    const float* cbr = (const float*)d_in[40]; const float* cbi = (const float*)d_in[41];

    // --- workspace layout (floats) ---
    float* ws = (float*)d_ws;
    constexpr size_t C1OUT = (size_t)B_ * C1_ * H1O * W1O;  // 14,909,440
    constexpr size_t P1SZ  = (size_t)B_ * C1_ * P1H * P1W;  //  3,727,360
    constexpr size_t C2OUT = (size_t)B_ * C2_ * H2O * W2O;  //  6,656,000
    constexpr size_t P2SZ  = (size_t)B_ * C2_ * P2H * P2W;  //  1,664,000  == [64,26000]
    constexpr size_t FCSZ  = (size_t)B_ * HID_;             //     32,768
    constexpr size_t CLSSZ = (size_t)B_ * NCLS_;            //        640
    constexpr int    SMAX  = 32;                            // max split-K factor
    constexpr size_t PSZ   = (size_t)SMAX * FCSZ;           //  1,048,576 per r/i
    size_t o = 0;
    float* c1r = ws + o; o += C1OUT;   float* c1i = ws + o; o += C1OUT;
    float* p1r = ws + o; o += P1SZ;    float* p1i = ws + o; o += P1SZ;
    float* c2r = ws + o; o += C2OUT;   float* c2i = ws + o; o += C2OUT;
    float* p2r = ws + o; o += P2SZ;    float* p2i = ws + o; o += P2SZ;
    float* f1r = ws + o; o += FCSZ;    float* f1ii = ws + o; o += FCSZ;
    float* f2r = ws + o; o += FCSZ;    float* f2ii = ws + o; o += FCSZ;
    float* clr = ws + o; o += CLSSZ;   float* cli  = ws + o; o += CLSSZ;
    float* PR  = ws + o; o += PSZ;     float* PI2  = ws + o; o += PSZ;
    float* stats = ws + o; o += 5 * 512;   // reused for each BN stage
    float* coef  = ws + o; o += 8 * 512;   // reused for each BN stage

    const int T = 256;
    float* outp = (float*)d_out;

    auto launch_cgemm = [&](const float* Xr, const float* Xi, const float* Wr, const float* Wi,
                            const float* bR, const float* bI, float* Yr, float* Yi,
                            int M, int N, int K, int S) {
        int KC = ((K + S - 1) / S + 7) & ~7;          // chunk, multiple of 8
        int tiles = ((M + 15) >> 4) * ((N + 15) >> 4);
        int waves = tiles * S;
        int blocks = (waves * 32 + T - 1) / T;
        cgemm_splitk_wmma_kernel<<<blocks, T, 0, stream>>>(Xr, Xi, Wr, Wi, PR, PI2,
                                                           M, N, K, S, KC);
        combine_kernel<<<(M * N + T - 1) / T, T, 0, stream>>>(PR, PI2, bR, bI, Yr, Yi, M, N, S);
    };

    // ---- stage 1: conv1 -> BN stats -> coef -> BN+ReLU+pool ----
    {
        int total = B_ * C1_ * H1O * W1O;
        conv1_kernel<<<(total + T - 1) / T, T, 0, stream>>>(x_re, x_im, w1r, w1i, b1r, b1i, c1r, c1i);
        zero_kernel<<<(5 * 512 + T - 1) / T, T, 0, stream>>>(stats, 5 * 512);
        moments_conv_kernel<<<dim3(C1_, 64), T, 0, stream>>>(c1r, c1i, C1_, H1O * W1O, stats);
        float invN = 1.0f / (float)((long)B_ * H1O * W1O);
        bncoef_kernel<<<1, 64, 0, stream>>>(stats, g1rr, g1ri, g1ii, be1r, be1i, invN, C1_, coef);
        int ptotal = B_ * C1_ * P1H * P1W;
        bnpool_kernel<<<(ptotal + T - 1) / T, T, 0, stream>>>(c1r, c1i, coef, C1_, H1O, W1O, p1r, p1i);
    }

    // ---- stage 2: conv2 -> BN stats -> coef -> BN+ReLU+pool ----
    {
        int total = B_ * C2_ * H2O * W2O;
        conv2_kernel<<<(total + T - 1) / T, T, 0, stream>>>(p1r, p1i, w2r, w2i, b2r, b2i, c2r, c2i);
        zero_kernel<<<(5 * 512 + T - 1) / T, T, 0, stream>>>(stats, 5 * 512);
        moments_conv_kernel<<<dim3(C2_, 64), T, 0, stream>>>(c2r, c2i, C2_, H2O * W2O, stats);
        float invN = 1.0f / (float)((long)B_ * H2O * W2O);
        bncoef_kernel<<<1, 64, 0, stream>>>(stats, g2rr, g2ri, g2ii, be2r, be2i, invN, C2_, coef);
        int ptotal = B_ * C2_ * P2H * P2W;
        bnpool_kernel<<<(ptotal + T - 1) / T, T, 0, stream>>>(c2r, c2i, coef, C2_, H2O, W2O, p2r, p2i);
    }

    // ---- stage 3: FC1 (split-K 32 -> 4096 waves) -> BN -> ReLU ----
    {
        launch_cgemm(p2r, p2i, f1wr, f1wi, f1br, f1bi, f1r, f1ii, B_, HID_, FLAT_, 32);
        moments_fc_kernel<<<(HID_ + 63) / 64, 64, 0, stream>>>(f1r, f1ii, HID_, stats);
        bncoef_kernel<<<(HID_ + 63) / 64, 64, 0, stream>>>(stats, g3rr, g3ri, g3ii, be3r, be3i,
                                                           1.0f / (float)B_, HID_, coef);
        bnrelu_fc_kernel<<<(B_ * HID_ + T - 1) / T, T, 0, stream>>>(f1r, f1ii, coef, HID_);
    }

    // ---- stage 4: FC2 (split-K 8 -> 1024 waves) -> BN -> ReLU ----
    {
        launch_cgemm(f1r, f1ii, f2wr, f2wi, f2br, f2bi, f2r, f2ii, B_, HID_, HID_, 8);
        moments_fc_kernel<<<(HID_ + 63) / 64, 64, 0, stream>>>(f2r, f2ii, HID_, stats);
        bncoef_kernel<<<(HID_ + 63) / 64, 64, 0, stream>>>(stats, g4rr, g4ri, g4ii, be4r, be4i,
                                                           1.0f / (float)B_, HID_, coef);
        bnrelu_fc_kernel<<<(B_ * HID_ + T - 1) / T, T, 0, stream>>>(f2r, f2ii, coef, HID_);
    }

    // ---- stage 5: classifier (N=10 masked, split-K 16) -> |h|^2 -> log_softmax ----
    {
        launch_cgemm(f2r, f2ii, cwr, cwi, cbr, cbi, clr, cli, B_, NCLS_, HID_, 16);
        logsoftmax_kernel<<<1, 64, 0, stream>>>(clr, cli, outp);
    }
}